// AffineSpatialTransformer_27058293965287
// MI455X (gfx1250) — compile-verified
//
#include <hip/hip_runtime.h>

typedef __attribute__((ext_vector_type(2))) float v2f;
typedef __attribute__((ext_vector_type(8))) float v8f;

#define N_  8
#define C_  2
#define D_  96
#define H_  160
#define W_  160
#define DHW_ (D_ * H_ * W_)

static __device__ __forceinline__ int clampi(int v, int lo, int hi) {
    return v < lo ? lo : (v > hi ? hi : v);
}

// ---------------------------------------------------------------------------
// Fold theta -> per-batch voxel-space affine coefficients, via WMMA f32.
//
//   xg = ix*kx - 1 (kx = 2/(W-1)), etc.;  u = theta_row . (xg,yg,zg,1)
//   coord = (u+1) * s   (s = (dim-1)/2)
// =>  coord_r = A[r] . (ix,iy,iz,1),  A = diag(s_r) * (Theta(24x4) * M + E3)
//   M = [[kx,0,0,-1],[0,ky,0,-1],[0,0,kz,-1],[0,0,0,1]],  E3: col3 += 1
//
// One wave, two V_WMMA_F32_16X16X4_F32 ops cover the 24 theta rows.
// ---------------------------------------------------------------------------
__global__ __launch_bounds__(32)
void fold_theta_wmma(const float* __restrict__ theta, float* __restrict__ coef)
{
    const int lane = threadIdx.x;       // 0..31
    const int l15  = lane & 15;         // A: row M ; B/C/D: column N
    const int half = lane >> 4;         // K half for A/B, M half for D
    const int k0   = half * 2;

    const float kx = 2.0f / (float)(W_ - 1);
    const float ky = 2.0f / (float)(H_ - 1);
    const float kz = 2.0f / (float)(D_ - 1);

    // B[k][n] = M[k][n] for n<4, else 0.  (lane = column n, vgpr+half = k)
    auto Mval = [&](int k, int nn) -> float {
        if (nn >= 4) return 0.0f;
        if (nn == 3) return (k == 3) ? 1.0f : -1.0f;
        if (nn == k) return (k == 0) ? kx : (k == 1) ? ky : (k == 2) ? kz : 0.0f;
        return 0.0f;
    };
    v2f b;
    b.x = Mval(k0, l15);
    b.y = Mval(k0 + 1, l15);

    // C[m][3] = 1 : the "+1" offset folded in before row scaling.
    const float cval = (l15 == 3) ? 1.0f : 0.0f;
    v8f c;
    for (int i = 0; i < 8; ++i) c[i] = cval;

    for (int base = 0; base < 24; base += 16) {
        // A: lane holds theta row (base + l15), K elements k0/k0+1.
        const int row    = base + l15;
        const int rc     = (row < 24) ? row : 23;
        const float mask = (row < 24) ? 1.0f : 0.0f;
        v2f a;
        a.x = theta[rc * 4 + k0] * mask;
        a.y = theta[rc * 4 + k0 + 1] * mask;

        // D = A x B + C  (16x16 f32 accumulator in 8 VGPRs)
        v8f d = __builtin_amdgcn_wmma_f32_16x16x4_f32(
            false, a, false, b, (short)0, c, false, false);

        // D layout: vgpr v, lanes 0-15 -> M=v, lanes 16-31 -> M=8+v; N = l15.
        if (l15 < 4) {
            for (int v = 0; v < 8; ++v) {
                const int m = v + 8 * half;
                const int r = base + m;
                if (r < 24) {
                    const int rr = r % 3;   // 0:x 1:y 2:z
                    const float s = (rr == 0) ? (float)(W_ - 1) * 0.5f
                                  : (rr == 1) ? (float)(H_ - 1) * 0.5f
                                              : (float)(D_ - 1) * 0.5f;
                    coef[r * 4 + l15] = d[v] * s;
                }
            }
        }
    }
}

// ---------------------------------------------------------------------------
// Main kernel: one block per output row (n,z,y), 160 threads along x.
// Memory-bound gather: 16 clamped global_load_b32 per thread (8 corners x 2
// channels, channels share coordinates/weights), zeros-padding via masked
// weights (branchless), nontemporal stores so the 157MB output stream does
// not evict the src working set (157MB, fits 192MB L2).
// ---------------------------------------------------------------------------
__global__ __launch_bounds__(W_)
void trilerp_affine(const float* __restrict__ src,
                    const float* __restrict__ coef,
                    float* __restrict__ out)
{
    const int bid = blockIdx.x;
    const int n   = bid / (D_ * H_);
    const int rzy = bid - n * (D_ * H_);
    const int z   = rzy / H_;
    const int y   = rzy - z * H_;
    const int x   = threadIdx.x;

    const float* __restrict__ A = coef + n * 12;    // block-uniform -> s_loads
    const float fx = (float)x, fy = (float)y, fz = (float)z;

    const float xs = fmaf(A[0], fx, fmaf(A[1], fy, fmaf(A[2],  fz, A[3])));
    const float ys = fmaf(A[4], fx, fmaf(A[5], fy, fmaf(A[6],  fz, A[7])));
    const float zs = fmaf(A[8], fx, fmaf(A[9], fy, fmaf(A[10], fz, A[11])));

    const float xf = floorf(xs), yf = floorf(ys), zf = floorf(zs);
    const float tx = xs - xf, ty = ys - yf, tz = zs - zf;
    const int x0 = (int)xf, y0 = (int)yf, z0 = (int)zf;
    const int x1 = x0 + 1, y1 = y0 + 1, z1 = z0 + 1;

    // zeros padding: OOB corner -> weight 0 (indices clamped so loads are safe)
    const float wx0 = ((unsigned)x0 < (unsigned)W_) ? (1.0f - tx) : 0.0f;
    const float wx1 = ((unsigned)x1 < (unsigned)W_) ? tx          : 0.0f;
    const float wy0 = ((unsigned)y0 < (unsigned)H_) ? (1.0f - ty) : 0.0f;
    const float wy1 = ((unsigned)y1 < (unsigned)H_) ? ty          : 0.0f;
    const float wz0 = ((unsigned)z0 < (unsigned)D_) ? (1.0f - tz) : 0.0f;
    const float wz1 = ((unsigned)z1 < (unsigned)D_) ? tz          : 0.0f;

    const int xc0 = clampi(x0, 0, W_ - 1), xc1 = clampi(x1, 0, W_ - 1);
    const int yc0 = clampi(y0, 0, H_ - 1), yc1 = clampi(y1, 0, H_ - 1);
    const int zc0 = clampi(z0, 0, D_ - 1), zc1 = clampi(z1, 0, D_ - 1);

    const int r00 = (zc0 * H_ + yc0) * W_;
    const int r01 = (zc0 * H_ + yc1) * W_;
    const int r10 = (zc1 * H_ + yc0) * W_;
    const int r11 = (zc1 * H_ + yc1) * W_;

    const int l0 = r00 + xc0, l1 = r00 + xc1;
    const int l2 = r01 + xc0, l3 = r01 + xc1;
    const int l4 = r10 + xc0, l5 = r10 + xc1;
    const int l6 = r11 + xc0, l7 = r11 + xc1;

    const float* __restrict__ s0 = src + (size_t)n * C_ * DHW_;   // channel 0
    const float* __restrict__ s1 = s0 + DHW_;                      // channel 1

    // Issue all 16 independent gathers up front for latency hiding.
    const float a0 = s0[l0], a1 = s0[l1], a2 = s0[l2], a3 = s0[l3];
    const float a4 = s0[l4], a5 = s0[l5], a6 = s0[l6], a7 = s0[l7];
    const float b0 = s1[l0], b1 = s1[l1], b2 = s1[l2], b3 = s1[l3];
    const float b4 = s1[l4], b5 = s1[l5], b6 = s1[l6], b7 = s1[l7];

    const float w00 = wz0 * wy0, w01 = wz0 * wy1;
    const float w10 = wz1 * wy0, w11 = wz1 * wy1;
    const float w000 = w00 * wx0, w001 = w00 * wx1;
    const float w010 = w01 * wx0, w011 = w01 * wx1;
    const float w100 = w10 * wx0, w101 = w10 * wx1;
    const float w110 = w11 * wx0, w111 = w11 * wx1;

    float acc0 = a0 * w000;
    float acc1 = b0 * w000;
    acc0 = fmaf(a1, w001, acc0);  acc1 = fmaf(b1, w001, acc1);
    acc0 = fmaf(a2, w010, acc0);  acc1 = fmaf(b2, w010, acc1);
    acc0 = fmaf(a3, w011, acc0);  acc1 = fmaf(b3, w011, acc1);
    acc0 = fmaf(a4, w100, acc0);  acc1 = fmaf(b4, w100, acc1);
    acc0 = fmaf(a5, w101, acc0);  acc1 = fmaf(b5, w101, acc1);
    acc0 = fmaf(a6, w110, acc0);  acc1 = fmaf(b6, w110, acc1);
    acc0 = fmaf(a7, w111, acc0);  acc1 = fmaf(b7, w111, acc1);

    const int lrow = (z * H_ + y) * W_ + x;
    float* o0 = out + (size_t)n * C_ * DHW_ + lrow;
    __builtin_nontemporal_store(acc0, o0);          // NT: keep L2 for src
    __builtin_nontemporal_store(acc1, o0 + DHW_);
}

extern "C" void kernel_launch(void* const* d_in, const int* in_sizes, int n_in,
                              void* d_out, int out_size, void* d_ws, size_t ws_size,
                              hipStream_t stream)
{
    (void)in_sizes; (void)n_in; (void)out_size; (void)ws_size;
    const float* src   = (const float*)d_in[0];   // [8,2,96,160,160] f32
    const float* theta = (const float*)d_in[1];   // [8,3,4] f32
    float* out  = (float*)d_out;                  // [8,2,96,160,160] f32
    float* coef = (float*)d_ws;                   // 96 floats of scratch

    // 1) Fold theta into voxel-space affine coefficients (WMMA f32, 1 wave).
    fold_theta_wmma<<<1, 32, 0, stream>>>(theta, coef);
    // 2) Bandwidth-bound trilinear gather: one block per (n,z,y) row.
    trilerp_affine<<<N_ * D_ * H_, W_, 0, stream>>>(src, coef, out);
}